// Dictionary_56212531970303
// MI455X (gfx1250) — compile-verified
//
#include <hip/hip_runtime.h>
#include <math.h>

// ---------------- problem constants ----------------
#define NB_ATOMS 64
#define NB_FREE  8
#define TAPS     81         // 9*9
#define HW       16384      // 128*128
#define BB       4
#define TOLV     1.0e-4f
#define N_ITERS  20

// ---------------- workspace layout (float offsets) ----------------
#define WS_GRAM   1024L      // 81*81 = 6561
#define WS_PF     8192L      // 6561
#define WS_PI     15360L     // 289
#define WS_SCAL   15872L     // [0]=beta [1]=L [2]=lmbda [3]=sc2, [4]=Lpk bits(uint)
#define WS_ATOMS  16384L     // 64*81 processed atoms (af3)
#define WS_G      26624L     // 64*160 combined GEMM matrix
#define WS_ACTIVE 36864L     // 4 ints
#define WS_RED    36896L     // partial reductions: d2[4*64], i2[4*64] = 512
#define WS_TEMPI  37440L     // 65536
#define WS_IMG    102976L    // 4 x 65536 (pair0 old,new, pair1 old,new)
#define WS_COEF   365120L    // 4 x 4194304

typedef __attribute__((ext_vector_type(2))) float v2f;
typedef __attribute__((ext_vector_type(8))) float v8f;

// =====================================================================
// P1: Bjorck orthonormalization of free atoms (8x81), gram, proj filters
// =====================================================================
__global__ __launch_bounds__(256) void k_bjorck(const float* __restrict__ fap,
                                                float* __restrict__ ws) {
  __shared__ float Wa[NB_FREE * TAPS];
  __shared__ float Wb[NB_FREE * TAPS];
  __shared__ float S[NB_FREE * NB_FREE];
  __shared__ float gramS[TAPS * TAPS];
  __shared__ float pm[TAPS];
  __shared__ float sInv;
  int t = threadIdx.x;

  for (int i = t; i < NB_FREE * TAPS; i += 256) Wa[i] = fap[i];
  if (t == 0) ((unsigned*)(ws + WS_SCAL))[4] = 0u;   // zero L_pk max bits
  __syncthreads();

  // S = W W^T  (8x8)
  if (t < 64) {
    int i = t >> 3, j = t & 7;
    float s = 0.f;
    for (int k = 0; k < TAPS; ++k) s += Wa[i * TAPS + k] * Wa[j * TAPS + k];
    S[t] = s;
  }
  __syncthreads();

  // spectral norm of W via power iteration on S (8x8)
  if (t == 0) {
    float v[8];
    for (int i = 0; i < 8; ++i) v[i] = 1.f;
    float lam = 0.f;
    for (int it = 0; it < 500; ++it) {
      float nv[8]; float nn = 0.f;
      for (int i = 0; i < 8; ++i) {
        float s = 0.f;
        for (int j = 0; j < 8; ++j) s += S[i * 8 + j] * v[j];
        nv[i] = s; nn += s * s;
      }
      nn = sqrtf(nn); lam = nn;
      float inv = (nn > 0.f) ? 1.f / nn : 0.f;
      for (int i = 0; i < 8; ++i) v[i] = nv[i] * inv;
    }
    float sigma = sqrtf(fmaxf(lam, 1e-30f));
    sInv = (sigma > 0.f) ? 1.f / sigma : 0.f;
  }
  __syncthreads();
  for (int i = t; i < NB_FREE * TAPS; i += 256) Wa[i] *= sInv;
  __syncthreads();

  // 15 Bjorck iterations: W = 1.5 W - 0.5 (W W^T) W
  for (int bj = 0; bj < 15; ++bj) {
    if (t < 64) {
      int i = t >> 3, j = t & 7;
      float s = 0.f;
      for (int k = 0; k < TAPS; ++k) s += Wa[i * TAPS + k] * Wa[j * TAPS + k];
      S[t] = s;
    }
    __syncthreads();
    for (int e = t; e < NB_FREE * TAPS; e += 256) {
      int i = e / TAPS, j = e % TAPS;
      float s = 0.f;
      for (int k = 0; k < 8; ++k) s += S[i * 8 + k] * Wa[k * TAPS + j];
      Wb[e] = 1.5f * Wa[e] - 0.5f * s;
    }
    __syncthreads();
    for (int e = t; e < NB_FREE * TAPS; e += 256) Wa[e] = Wb[e];
    __syncthreads();
  }

  // gram = fa^T fa (81x81)
  float* gramWS = ws + WS_GRAM;
  for (int e = t; e < TAPS * TAPS; e += 256) {
    int i = e / TAPS, j = e % TAPS;
    float s = 0.f;
    for (int k = 0; k < 8; ++k) s += Wa[k * TAPS + i] * Wa[k * TAPS + j];
    gramS[e] = s; gramWS[e] = s;
  }
  __syncthreads();
  // pm[j] = mean_k proj[j,k] = (1 - rowsum_j(gram)) / 81
  if (t < TAPS) {
    float s = 0.f;
    for (int k = 0; k < TAPS; ++k) s += gramS[t * TAPS + k];
    pm[t] = (1.f - s) / 81.f;
  }
  __syncthreads();
  // proj_filters[i][j] = (I - gram)[i][j] - pm[j]
  float* pfWS = ws + WS_PF;
  for (int e = t; e < TAPS * TAPS; e += 256) {
    int i = e / TAPS, j = e % TAPS;
    float pij = ((i == j) ? 1.f : 0.f) - gramS[e];
    pfWS[e] = pij - pm[j];
  }
}

// =====================================================================
// P2: proj_impulse (17x17) = conv_t(conv(dirac, pf), pf), circular on 17x17
// =====================================================================
__global__ __launch_bounds__(512) void k_projimpulse(float* __restrict__ ws) {
  const float* pf = ws + WS_PF;
  float* pi = ws + WS_PI;
  int t = threadIdx.x;
  if (t >= 289) return;
  int h = t / 17, w = t % 17;
  float s = 0.f;
  for (int dy = 0; dy < 9; ++dy) {
    for (int dx = 0; dx < 9; ++dx) {
      int hh = (h + dy - 4 + 17) % 17;
      int ww = (w + dx - 4 + 17) % 17;
      int dyy = (12 - hh + 17) % 17;
      int dxx = (12 - ww + 17) % 17;
      if (dyy < 9 && dxx < 9) {
        int ta = (8 - dy) * 9 + (8 - dx);
        int tb = dyy * 9 + dxx;
        float acc = 0.f;
        for (int c = 0; c < TAPS; ++c) acc += pf[c * TAPS + ta] * pf[c * TAPS + tb];
        s += acc;
      }
    }
  }
  pi[t] = s;
}

// =====================================================================
// P3: L_pk = max |DFT_128x128(zero-padded proj_impulse)|
// =====================================================================
__global__ __launch_bounds__(256) void k_dft_max(float* __restrict__ ws) {
  const float* pi = ws + WS_PI;
  unsigned* bits = (unsigned*)(ws + WS_SCAL) + 4;
  int n = blockIdx.x * blockDim.x + threadIdx.x;   // 0..16383
  int u = n >> 7, v = n & 127;
  const float w0 = -6.283185307179586f / 128.f;
  float re = 0.f, im = 0.f;
  for (int y = 0; y < 17; ++y) {
    float rr = 0.f, ri = 0.f;
    for (int x = 0; x < 17; ++x) {
      float sn, cs;
      __sincosf(w0 * (float)(v * x), &sn, &cs);
      float p = pi[y * 17 + x];
      rr += p * cs; ri += p * sn;
    }
    float s2, c2;
    __sincosf(w0 * (float)(u * y), &s2, &c2);
    re += rr * c2 - ri * s2;
    im += rr * s2 + ri * c2;
  }
  float mag = sqrtf(re * re + im * im);
  atomicMax(bits, __float_as_uint(mag));   // mags >= 0: uint order == float order
}

// =====================================================================
// P4: scalars  beta, L, lmbda, sc2
// =====================================================================
__global__ void k_scalars(const float* __restrict__ beta_p,
                          const float* __restrict__ lmbda_p,
                          float* __restrict__ ws) {
  float* sc = ws + WS_SCAL;
  float beta = fmaxf(beta_p[0], 0.f);
  float lpk = __uint_as_float(((unsigned*)sc)[4]);
  sc[0] = beta;
  sc[1] = 1.01f * beta * lpk;
  sc[2] = lmbda_p[0];
  sc[3] = sqrtf(0.99f / fmaxf(beta, 0.1f));
}

// =====================================================================
// P5: atoms processing -> af3 (64x81), Xmat, combined G (64x160)
// =====================================================================
__global__ __launch_bounds__(256) void k_atoms(const float* __restrict__ atoms_p,
                                               float* __restrict__ ws) {
  __shared__ float af[NB_ATOMS * TAPS];
  __shared__ float af2[NB_ATOMS * TAPS];
  __shared__ float uu[NB_ATOMS];
  __shared__ float vv[TAPS];
  __shared__ float wv[TAPS];
  __shared__ float red0;
  const float* gram = ws + WS_GRAM;
  const float* sc = ws + WS_SCAL;
  float* aout = ws + WS_ATOMS;
  float* G = ws + WS_G;
  int t = threadIdx.x;

  for (int i = t; i < NB_ATOMS * TAPS; i += 256) af[i] = atoms_p[i];
  __syncthreads();
  if (t < NB_ATOMS) {
    float s = 0.f;
    for (int j = 0; j < TAPS; ++j) s += af[t * TAPS + j];
    uu[t] = s / 81.f;
  }
  __syncthreads();
  for (int i = t; i < NB_ATOMS * TAPS; i += 256) af[i] -= uu[i / TAPS];
  __syncthreads();
  // af2 = af - af @ gram
  for (int e = t; e < NB_ATOMS * TAPS; e += 256) {
    int o = e / TAPS, j = e % TAPS;
    float s = 0.f;
    for (int k = 0; k < TAPS; ++k) s += af[o * TAPS + k] * gram[k * TAPS + j];
    af2[e] = af[e] - s;
  }
  __syncthreads();
  // row-normalize
  if (t < NB_ATOMS) {
    float s = 0.f;
    for (int j = 0; j < TAPS; ++j) { float x = af2[t * TAPS + j]; s += x * x; }
    uu[t] = rsqrtf(fmaxf(s, 1e-30f));
  }
  __syncthreads();
  for (int e = t; e < NB_ATOMS * TAPS; e += 256) af2[e] *= uu[e / TAPS];
  __syncthreads();
  // spectral norm of af2 via power iteration on M = af2^T af2 (81x81)
  if (t < TAPS) vv[t] = 1.f / 9.f;
  __syncthreads();
  float lam = 0.f;
  for (int it = 0; it < 300; ++it) {
    if (t < NB_ATOMS) {
      float s = 0.f;
      for (int j = 0; j < TAPS; ++j) s += af2[t * TAPS + j] * vv[j];
      uu[t] = s;
    }
    __syncthreads();
    if (t < TAPS) {
      float s = 0.f;
      for (int o = 0; o < NB_ATOMS; ++o) s += af2[o * TAPS + t] * uu[o];
      wv[t] = s;
    }
    __syncthreads();
    if (t == 0) {
      float s = 0.f;
      for (int j = 0; j < TAPS; ++j) s += wv[j] * wv[j];
      red0 = sqrtf(s);
    }
    __syncthreads();
    lam = red0;                                // -> lambda_max(M) = sigma^2
    float inv = (lam > 0.f) ? 1.f / lam : 0.f;
    if (t < TAPS) vv[t] = wv[t] * inv;
    __syncthreads();
  }
  float scale = sc[3] / sqrtf(fmaxf(lam, 1e-30f));   // sc2 / sigma
  for (int e = t; e < NB_ATOMS * TAPS; e += 256) af2[e] *= scale;
  __syncthreads();
  for (int e = t; e < NB_ATOMS * TAPS; e += 256) aout[e] = af2[e];
  // G[:, 0:64] = Xmat = af3 af3^T ; G[:, 64:145] = -af3 ; rest 0
  for (int e = t; e < NB_ATOMS * NB_ATOMS; e += 256) {
    int o = e >> 6, p = e & 63;
    float s = 0.f;
    for (int j = 0; j < TAPS; ++j) s += af2[o * TAPS + j] * af2[p * TAPS + j];
    G[o * 160 + p] = s;
  }
  for (int e = t; e < NB_ATOMS * 96; e += 256) {
    int o = e / 96, jj = e % 96;
    G[o * 160 + 64 + jj] = (jj < TAPS) ? -af2[o * TAPS + jj] : 0.f;
  }
}

// =====================================================================
// Init: imgs pair0 = y, coef pair0 = 0, active = 1
// =====================================================================
__global__ void k_init(const float* __restrict__ y, float* __restrict__ ws) {
  long gid = (long)blockIdx.x * blockDim.x + threadIdx.x;
  long stride = (long)gridDim.x * blockDim.x;
  float* c0 = ws + WS_COEF;
  for (long i = gid; i < 2L * BB * NB_ATOMS * HW; i += stride) c0[i] = 0.f;
  float* im0 = ws + WS_IMG;
  for (long i = gid; i < (long)BB * HW; i += stride) {
    float v = y[i];
    im0[i] = v;
    im0[(long)BB * HW + i] = v;
  }
  if (gid < BB) ((int*)(ws + WS_ACTIVE))[gid] = 1;
}

// =====================================================================
// K1: coefficient update, fused WMMA GEMM:
//   grad = Xmat @ temp_c - conv(new_img, atoms)  as  G (64x160) @ Bcomb
// Block = 64 consecutive pixels of one image row (4 waves x 16 pixels).
// LDS: combined G (40KB) + 9 wrapped image rows (4.5KB).
// Fully unrolled K: 16 temp_c steps + 21 conv-tap steps, no divergence;
// invalid taps (t>=81) read tap 0 and hit zero-padded G columns.
// =====================================================================
__global__ __launch_bounds__(128) void k_coeff(
    const float* __restrict__ cO, const float* __restrict__ cN,
    const float* __restrict__ nimg, const float* __restrict__ G,
    const float* __restrict__ scal, const int* __restrict__ act,
    float* __restrict__ cOo, float* __restrict__ cNo, float mom) {
  __shared__ float sG[64 * 160];
  __shared__ float sImg[9 * 128];
  int tid = threadIdx.x;

  int gpix0 = blockIdx.x << 6;          // 64 pixels per block, within B*HW
  int b = gpix0 >> 14;
  int prow = (gpix0 & 16383) >> 7;      // image row (same for whole block)
  int w0 = gpix0 & 127;                 // 0 or 64
  const float* imb = nimg + (b << 14);

  for (int i = tid; i < 64 * 160; i += 128) sG[i] = G[i];
  for (int i = tid; i < 9 * 128; i += 128) {
    int r = i >> 7, c = i & 127;
    sImg[i] = imb[(((prow + r - 4) & 127) << 7) + c];
  }
  __syncthreads();

  int wv = tid >> 5, lane = tid & 31;
  int hs = lane >> 4;                   // half-wave: K offset +2
  int l16 = lane & 15;
  int w = w0 + (wv << 4) + l16;
  int pix = (prow << 7) + w;

  const float* cOb = cO + ((long)b << 20);   // 64*HW = 1<<20
  const float* cNb = cN + ((long)b << 20);
  float beta = scal[0], lmbda = scal[2];
  int a = act[b];

  // wrapped column indices for the 9 conv taps of this lane
  int wcol[9];
#pragma unroll
  for (int dx = 0; dx < 9; ++dx) wcol[dx] = (w + dx - 4) & 127;

  v8f zero8 = {0.f, 0.f, 0.f, 0.f, 0.f, 0.f, 0.f, 0.f};
  v8f acc[4];
  acc[0] = zero8; acc[1] = zero8; acc[2] = zero8; acc[3] = zero8;

  // ---- phase 1: K = 0..63, temp_c rows; base ptr + immediate offsets ----
  const float* pO = cOb + pix + (hs << 15);   // + hs*2*HW
  const float* pN = cNb + pix + (hs << 15);
#pragma unroll
  for (int ks = 0; ks < 16; ++ks) {
    long off = (long)ks << 16;                // ks*4*HW elements
    float o0 = pO[off], n0 = pN[off];
    float o1 = pO[off + HW], n1 = pN[off + HW];
    v2f bop;
    bop.x = n0 + mom * (n0 - o0);
    bop.y = n1 + mom * (n1 - o1);
    int k0 = (ks << 2) + (hs << 1);
#pragma unroll
    for (int mt = 0; mt < 4; ++mt) {
      int row = (mt << 4) + l16;
      v2f aop;
      aop.x = sG[row * 160 + k0];
      aop.y = sG[row * 160 + k0 + 1];
      acc[mt] = __builtin_amdgcn_wmma_f32_16x16x4_f32(
          false, aop, false, bop, (short)0, acc[mt], false, false);
    }
  }

  // ---- phase 2: K = 64..147, conv taps from LDS image slab ----
#pragma unroll
  for (int j = 0; j < 21; ++j) {
    int tA0 = 4 * j, tA1 = 4 * j + 1;         // taps for hs==0
    int tB0 = 4 * j + 2, tB1 = 4 * j + 3;     // taps for hs==1
    if (tA1 > 80) tA1 = 0;                    // clamp: zero G column anyway
    if (tB0 > 80) tB0 = 0;
    if (tB1 > 80) tB1 = 0;
    int r0 = hs ? (tB0 / 9) : (tA0 / 9);      // compile-time constants + cndmask
    int c0 = hs ? (tB0 % 9) : (tA0 % 9);
    int r1 = hs ? (tB1 / 9) : (tA1 / 9);
    int c1 = hs ? (tB1 % 9) : (tA1 % 9);
    v2f bop;
    bop.x = sImg[(r0 << 7) + wcol[c0]];
    bop.y = sImg[(r1 << 7) + wcol[c1]];
    int k0 = 64 + (j << 2) + (hs << 1);
#pragma unroll
    for (int mt = 0; mt < 4; ++mt) {
      int row = (mt << 4) + l16;
      v2f aop;
      aop.x = sG[row * 160 + k0];
      aop.y = sG[row * 160 + k0 + 1];
      acc[mt] = __builtin_amdgcn_wmma_f32_16x16x4_f32(
          false, aop, false, bop, (short)0, acc[mt], false, false);
    }
  }

  // ---- epilogue: soft-threshold + masked ping-pong writes ----
  long ebase = ((long)b << 20) + pix + ((long)hs << 17);   // + hs*8*HW
  const float* eO = cO + ebase;
  const float* eN = cN + ebase;
  float* eOo = cOo + ebase;
  float* eNo = cNo + ebase;
#pragma unroll
  for (int mt = 0; mt < 4; ++mt) {
#pragma unroll
    for (int r = 0; r < 8; ++r) {
      long off = ((long)((mt << 4) + r)) << 14;   // constant per unrolled iter
      float oc = eO[off], nc = eN[off];
      float tc = nc + mom * (nc - oc);
      float g = acc[mt][r];
      float uv = tc - beta * g;
      float st = copysignf(fmaxf(fabsf(uv) - lmbda, 0.f), uv);
      eOo[off] = a ? nc : oc;
      eNo[off] = a ? st : nc;
    }
  }
}

// =====================================================================
// K2a: temp_i and img2 (elementwise)
// =====================================================================
__global__ void k_imgA(const float* __restrict__ img, const float* __restrict__ nimg,
                       const int* __restrict__ act, float* __restrict__ tempi,
                       float* __restrict__ imgOutOld, float mom) {
  int i = blockIdx.x * blockDim.x + threadIdx.x;   // B*HW threads
  int b = i >> 14;
  float a0 = img[i], n0 = nimg[i];
  tempi[i] = n0 + mom * (n0 - a0);
  imgOutOld[i] = act[b] ? n0 : a0;
}

// =====================================================================
// K2b: dict_pred (64-ch 9x9 transpose conv) + 17x17 proj conv + img update
// LDS-tiled 16x16 spatial tile per block; deterministic partial reduction.
// =====================================================================
__global__ __launch_bounds__(256) void k_imgB(
    const float* __restrict__ img, const float* __restrict__ nimg,
    const float* __restrict__ y, const float* __restrict__ tempi,
    const float* __restrict__ cnew2, const float* __restrict__ at3,
    const float* __restrict__ piW, const float* __restrict__ scal,
    const int* __restrict__ act, float* __restrict__ imgOutNew,
    float* __restrict__ redD, float* __restrict__ redI) {
  __shared__ float sAt[NB_ATOMS * TAPS];
  __shared__ float sPi[289];
  __shared__ float ch[24 * 24];
  __shared__ float th[32 * 32];
  __shared__ float rd[256], ri[256];
  int t = threadIdx.x;
  int b = blockIdx.z;
  int ty0 = blockIdx.y << 4, tx0 = blockIdx.x << 4;
  int ly = t >> 4, lx = t & 15;
  int h = ty0 + ly, w = tx0 + lx;

  for (int i = t; i < NB_ATOMS * TAPS; i += 256) sAt[i] = at3[i];
  for (int i = t; i < 289; i += 256) sPi[i] = piW[i];

  const float* cb = cnew2 + ((long)b << 20);
  float dp = 0.f;
  for (int o = 0; o < NB_ATOMS; ++o) {
    __syncthreads();
    for (int i = t; i < 576; i += 256) {
      int r = i / 24, c = i % 24;
      ch[i] = cb[((long)o << 14) + (((ty0 - 4 + r) & 127) << 7) + ((tx0 - 4 + c) & 127)];
    }
    __syncthreads();
    const float* ao = sAt + o * TAPS;
    for (int ey = 0; ey < 9; ++ey)
      for (int ex = 0; ex < 9; ++ex)
        dp += ao[ey * 9 + ex] * ch[(ly + 8 - ey) * 24 + (lx + 8 - ex)];
  }
  __syncthreads();
  const float* tb = tempi + (b << 14);
  for (int i = t; i < 1024; i += 256) {
    int r = i >> 5, c = i & 31;
    th[i] = tb[(((ty0 - 8 + r) & 127) << 7) + ((tx0 - 8 + c) & 127)];
  }
  __syncthreads();
  float sp = 0.f;
  for (int dy = 0; dy < 17; ++dy)
    for (int dx = 0; dx < 17; ++dx)
      sp += sPi[dy * 17 + dx] * th[(ly + dy) * 32 + (lx + dx)];

  int pix = (h << 7) + w;
  long gi = ((long)b << 14) + pix;
  float ti = tb[pix];
  float beta = scal[0], L = scal[1];
  float upd = ti - (ti - y[gi] + beta * (sp - dp)) / L;
  int a = act[b];
  float oi = img[gi], ni = nimg[gi];
  float img2 = a ? ni : oi;
  float nimg2 = a ? upd : ni;
  imgOutNew[gi] = nimg2;
  float d = img2 - nimg2;
  rd[t] = d * d; ri[t] = img2 * img2;
  __syncthreads();
  for (int s = 128; s > 0; s >>= 1) {
    if (t < s) { rd[t] += rd[t + s]; ri[t] += ri[t + s]; }
    __syncthreads();
  }
  if (t == 0) {
    int blk = blockIdx.y * 8 + blockIdx.x;           // 64 blocks per batch
    redD[b * 64 + blk] = rd[0];
    redI[b * 64 + blk] = ri[0];
  }
}

// =====================================================================
// K3: per-batch residual -> active flags (deterministic fixed-order sum)
// =====================================================================
__global__ void k_res(float* __restrict__ ws) {
  int b = threadIdx.x;
  if (b < BB) {
    const float* redD = ws + WS_RED;
    const float* redI = ws + WS_RED + BB * 64;
    float d2 = 0.f, i2 = 0.f;
    for (int k = 0; k < 64; ++k) { d2 += redD[b * 64 + k]; i2 += redI[b * 64 + k]; }
    ((int*)(ws + WS_ACTIVE))[b] = (d2 > TOLV * TOLV * i2) ? 1 : 0;
  }
}

// =====================================================================
// Output copy: d_out = [new_img (65536) | new_coeffs (4194304)]
// =====================================================================
__global__ void k_out(const float* __restrict__ ws, float* __restrict__ out) {
  long i = (long)blockIdx.x * blockDim.x + threadIdx.x;
  if (i < (long)BB * HW) {
    out[i] = ws[WS_IMG + (long)BB * HW + i];                  // pair0 new_img
  } else {
    out[i] = ws[WS_COEF + (long)BB * NB_ATOMS * HW + (i - (long)BB * HW)]; // pair0 c_new
  }
}

// =====================================================================
extern "C" void kernel_launch(void* const* d_in, const int* in_sizes, int n_in,
                              void* d_out, int out_size, void* d_ws, size_t ws_size,
                              hipStream_t stream) {
  (void)in_sizes; (void)n_in; (void)out_size; (void)ws_size;
  const float* y       = (const float*)d_in[0];   // (4,1,128,128)
  const float* atoms   = (const float*)d_in[1];   // (64,1,9,9)
  const float* fatoms  = (const float*)d_in[2];   // (8,1,9,9)
  const float* beta_p  = (const float*)d_in[3];
  const float* lmbda_p = (const float*)d_in[4];
  float* wsf = (float*)d_ws;
  float* outf = (float*)d_out;

  // ---- precompute ----
  k_bjorck<<<1, 256, 0, stream>>>(fatoms, wsf);
  k_projimpulse<<<1, 512, 0, stream>>>(wsf);
  k_dft_max<<<64, 256, 0, stream>>>(wsf);
  k_scalars<<<1, 1, 0, stream>>>(beta_p, lmbda_p, wsf);
  k_atoms<<<1, 256, 0, stream>>>(atoms, wsf);
  k_init<<<2048, 256, 0, stream>>>(y, wsf);

  float* imgP[2][2] = {
      {wsf + WS_IMG,                       wsf + WS_IMG + (long)BB * HW},
      {wsf + WS_IMG + 2L * BB * HW,        wsf + WS_IMG + 3L * BB * HW}};
  float* cofP[2][2] = {
      {wsf + WS_COEF,                          wsf + WS_COEF + (long)BB * NB_ATOMS * HW},
      {wsf + WS_COEF + 2L * BB * NB_ATOMS * HW, wsf + WS_COEF + 3L * BB * NB_ATOMS * HW}};
  int* act = (int*)(wsf + WS_ACTIVE);
  float* redD = wsf + WS_RED;
  float* redI = wsf + WS_RED + BB * 64;

  for (int it = 0; it < N_ITERS; ++it) {
    int pin = it & 1, pout = pin ^ 1;
    float mom = (float)it / (float)(it + 3);   // k = it+1: (k-1)/(k+2)
    k_coeff<<<1024, 128, 0, stream>>>(cofP[pin][0], cofP[pin][1], imgP[pin][1],
                                      wsf + WS_G, wsf + WS_SCAL, act,
                                      cofP[pout][0], cofP[pout][1], mom);
    k_imgA<<<256, 256, 0, stream>>>(imgP[pin][0], imgP[pin][1], act,
                                    wsf + WS_TEMPI, imgP[pout][0], mom);
    k_imgB<<<dim3(8, 8, BB), 256, 0, stream>>>(imgP[pin][0], imgP[pin][1], y,
                                               wsf + WS_TEMPI, cofP[pout][1],
                                               wsf + WS_ATOMS, wsf + WS_PI,
                                               wsf + WS_SCAL, act, imgP[pout][1],
                                               redD, redI);
    k_res<<<1, 32, 0, stream>>>(wsf);
  }
  // after 20 iterations final state lives in pair 0
  k_out<<<16640, 256, 0, stream>>>(wsf, outf);
}